// Self_Correlation_Per_52089363366002
// MI455X (gfx1250) — compile-verified
//
#include <hip/hip_runtime.h>
#include <math.h>

// Problem constants (from reference)
#define BATCH   4
#define CCH     256          // channels (K of the Gram GEMM)
#define HW      4096         // 64*64 pixels
#define KTOP    115          // NB_POOLS
#define EPSN    1e-12f

// Tiling
#define JB      64           // Gram columns per workgroup
#define IB      128          // Gram rows per iteration (8 waves x 16)
#define CST     65           // LDS C-stage row stride (floats)
#define NKS     (CCH / 4)    // 64 k-steps of 4

typedef __attribute__((ext_vector_type(2))) float v2f;
typedef __attribute__((ext_vector_type(8))) float v8f;

// ---------------------------------------------------------------------------
// Kernel 1: L2-normalize over channel dim; emit channel-major (B operand) and
// pixel-major (A operand, pre-scaled by 1/C) copies into workspace.
// ---------------------------------------------------------------------------
__global__ void sc_normalize_kernel(const float* __restrict__ x,
                                    float* __restrict__ xn_c,   // [B][C][HW]
                                    float* __restrict__ xn_t) { // [B][HW][C], *1/C
    const int idx = blockIdx.x * blockDim.x + threadIdx.x;   // 0 .. B*HW-1
    const int b = idx >> 12;           // /4096
    const int p = idx & (HW - 1);
    const float* px = x + (size_t)b * CCH * HW + p;

    float ss = 0.0f;
    #pragma unroll 8
    for (int c = 0; c < CCH; ++c) {
        float v = px[(size_t)c * HW];
        ss += v * v;
    }
    const float scale  = 1.0f / fmaxf(sqrtf(ss), EPSN);
    const float scaleA = scale * (1.0f / (float)CCH);  // fold Gram 1/C into A

    float* outc = xn_c + (size_t)b * CCH * HW + p;
    float* outt = xn_t + ((size_t)b * HW + p) * CCH;
    #pragma unroll 8
    for (int c = 0; c < CCH; ++c) {
        float v = px[(size_t)c * HW];
        outc[(size_t)c * HW] = v * scale;
        outt[c] = v * scaleA;
    }
}

// ---------------------------------------------------------------------------
// Min-heap helpers (115-entry heap per column, in LDS)
// ---------------------------------------------------------------------------
__device__ __forceinline__ void sc_sift_down(float* hp, int size) {
    int i = 0;
    float v = hp[0];
    for (;;) {
        int l = 2 * i + 1;
        if (l >= size) break;
        int s = l;
        float sv = hp[l];
        int r = l + 1;
        if (r < size) {
            float rv = hp[r];
            if (rv < sv) { s = r; sv = rv; }
        }
        if (sv >= v) break;
        hp[i] = sv;
        i = s;
    }
    hp[i] = v;
}

// ---------------------------------------------------------------------------
// Kernel 2: fused Gram-GEMM (fp32 WMMA 16x16x4) + streaming top-115 per column.
// Grid: BATCH * (HW/JB) blocks of 256 threads (8 wave32).
// B tile stored fragment-packed: BtF[(ks*4+jt)*32 + lane] = {B[4ks+lhi][n],
// B[4ks+lhi+2][n]} with n = jt*16 + l16 -> one ds_load_b64 per fragment,
// immediate offsets, all 64 banks hit, zero repack moves.
// ---------------------------------------------------------------------------
__global__ void __launch_bounds__(256)
sc_corr_topk_kernel(const float* __restrict__ xn_c,   // [B][C][HW]
                    const float* __restrict__ xn_t,   // [B][HW][C] (pre /C)
                    float* __restrict__ out) {        // [B][KTOP][HW]
    __shared__ v2f   BtF[NKS * 4 * 32];    // 64 KB  : fragment-packed B tile
    __shared__ float Ct[IB * CST];         // 32.5 KB: staged corr tile
    __shared__ float Hp[JB * KTOP];        // 28.75KB: per-column min-heaps

    const int bb = blockIdx.x >> 6;               // batch
    const int j0 = (blockIdx.x & 63) * JB;        // first Gram column
    const int tid  = threadIdx.x;
    const int lane = tid & 31;
    const int wv   = tid >> 5;                    // wave 0..7
    const int l16  = lane & 15;
    const int lhi  = lane >> 4;                   // 0 or 1

    // ---- Stage B tile into fragment-packed layout --------------------------
    {
        const int jj = tid & 63;                  // source column 0..63
        const int kb = tid >> 6;                  // k residue 0..3
        const int jt   = jj >> 4;
        const int l16j = jj & 15;
        const int laneD = ((kb & 1) << 4) + l16j; // dest lane
        const int half  = kb >> 1;                // .x or .y
        const float* src = xn_c + (size_t)bb * CCH * HW + j0 + jj;
        float* dstf = (float*)BtF;
        #pragma unroll 4
        for (int kk = 0; kk < NKS; ++kk) {
            const int k = kb + (kk << 2);
            dstf[(((kk << 2) + jt) * 32 + laneD) * 2 + half] = src[(size_t)k * HW];
        }
    }
    // ---- Init heaps ---------------------------------------------------------
    if (tid < JB) {
        float* hp = &Hp[tid * KTOP];
        for (int i = 0; i < KTOP; ++i) hp[i] = -INFINITY;
    }
    __syncthreads();

    // ---- Main loop over Gram rows ------------------------------------------
    for (int i0 = 0; i0 < HW; i0 += IB) {
        // Each wave: rows [i0 + 16*wv, +16), all 64 columns (4 WMMA tiles).
        v8f acc[4];
        #pragma unroll
        for (int jt = 0; jt < 4; ++jt) acc[jt] = (v8f){};

        // A fragment base: lane holds row m = i0+16*wv+l16, K pair at 2*lhi.
        const float* Ar =
            xn_t + ((size_t)bb * HW + i0 + wv * 16 + l16) * CCH + 2 * lhi;
        const v2f* Bf = BtF + lane;               // per-lane fragment base

        #pragma unroll 4
        for (int ks = 0; ks < NKS; ++ks) {
            const v2f a = *(const v2f*)(Ar + 4 * ks);   // A[m][k], A[m][k+1]
            #pragma unroll
            for (int jt = 0; jt < 4; ++jt) {
                const v2f bfr = Bf[((ks << 2) + jt) * 32];  // single ds_load_b64
                acc[jt] = __builtin_amdgcn_wmma_f32_16x16x4_f32(
                    /*neg_a=*/false, a, /*neg_b=*/false, bfr,
                    /*c_mod=*/(short)0, acc[jt],
                    /*reuse_a=*/false, /*reuse_b=*/false);
            }
        }

        // Stage tile (already scaled via A): C/D layout -> M = r + 8*lhi, N = l16.
        #pragma unroll
        for (int jt = 0; jt < 4; ++jt) {
            #pragma unroll
            for (int r = 0; r < 8; ++r) {
                Ct[(wv * 16 + r + 8 * lhi) * CST + jt * 16 + l16] = acc[jt][r];
            }
        }
        __syncthreads();

        // Streaming top-k update: one thread per column.
        if (tid < JB) {
            float* hp = &Hp[tid * KTOP];
            float mn = hp[0];
            for (int r = 0; r < IB; ++r) {
                const float v = Ct[r * CST + tid];
                if (v > mn) {
                    hp[0] = v;
                    sc_sift_down(hp, KTOP);
                    mn = hp[0];
                }
            }
        }
        __syncthreads();
    }

    // ---- Heapsort (min-heap extract-to-end => descending) + write ----------
    if (tid < JB) {
        float* hp = &Hp[tid * KTOP];
        for (int sz = KTOP; sz > 1; --sz) {
            const float mn = hp[0];
            hp[0] = hp[sz - 1];
            hp[sz - 1] = mn;
            sc_sift_down(hp, sz - 1);
        }
        const int j = j0 + tid;
        float* dst = out + ((size_t)bb * KTOP) * HW + j;
        for (int kidx = 0; kidx < KTOP; ++kidx) {
            dst[(size_t)kidx * HW] = hp[kidx];
        }
    }
}

// ---------------------------------------------------------------------------
extern "C" void kernel_launch(void* const* d_in, const int* in_sizes, int n_in,
                              void* d_out, int out_size, void* d_ws, size_t ws_size,
                              hipStream_t stream) {
    (void)in_sizes; (void)n_in; (void)out_size; (void)ws_size;
    const float* x = (const float*)d_in[0];
    float* out = (float*)d_out;

    float* xn_c = (float*)d_ws;                          // [B][C][HW]
    float* xn_t = xn_c + (size_t)BATCH * CCH * HW;       // [B][HW][C]

    sc_normalize_kernel<<<(BATCH * HW) / 256, 256, 0, stream>>>(x, xn_c, xn_t);
    sc_corr_topk_kernel<<<BATCH * (HW / JB), 256, 0, stream>>>(xn_c, xn_t, out);
}